// PolicyNet_69011534512790
// MI455X (gfx1250) — compile-verified
//
#include <hip/hip_runtime.h>
#include <math.h>

// ---------------------------------------------------------------------------
// Types for CDNA5 WMMA f32 16x16x4  (A: 16x4 -> 2 VGPR, B: 4x16 -> 2 VGPR,
// C/D: 16x16 -> 8 VGPR)
// ---------------------------------------------------------------------------
typedef float v2f __attribute__((ext_vector_type(2)));
typedef float v8f __attribute__((ext_vector_type(8)));

__device__ __forceinline__ v8f wmma_f32_4(v2f a, v2f b, v8f c) {
  // 8 args: (neg_a, A, neg_b, B, c_mod, C, reuse_a, reuse_b)
  return __builtin_amdgcn_wmma_f32_16x16x4_f32(false, a, false, b, (short)0, c,
                                               false, false);
}

// ---------------------------------------------------------------------------
// 1) CSR build: degree histogram, exclusive scan, scatter
// ---------------------------------------------------------------------------
__global__ void k_count_deg(const int* __restrict__ ei, int* __restrict__ deg,
                            int E) {
  int e = blockIdx.x * blockDim.x + threadIdx.x;
  if (e < E) atomicAdd(&deg[ei[E + e]], 1);  // row 1 = dst
}

__global__ __launch_bounds__(1024) void k_scan(const int* __restrict__ deg,
                                               int* __restrict__ offs, int n) {
  __shared__ int part[1024];
  int t = threadIdx.x;
  int chunk = (n + 1023) / 1024;
  int b0 = t * chunk;
  int b1 = min(n, b0 + chunk);
  int s = 0;
  for (int i = b0; i < b1; ++i) s += deg[i];
  part[t] = s;
  __syncthreads();
  for (int o = 1; o < 1024; o <<= 1) {
    int v = 0;
    if (t >= o) v = part[t - o];
    __syncthreads();
    if (t >= o) part[t] += v;
    __syncthreads();
  }
  int run = (t == 0) ? 0 : part[t - 1];
  for (int i = b0; i < b1; ++i) {
    offs[i] = run;
    run += deg[i];
  }
  if (t == 1023) offs[n] = part[1023];
}

__global__ void k_scatter(const int* __restrict__ ei,
                          const int* __restrict__ offs, int* __restrict__ cur,
                          int* __restrict__ csr_src, int E) {
  int e = blockIdx.x * blockDim.x + threadIdx.x;
  if (e >= E) return;
  int d = ei[E + e];
  int s = ei[e];
  int pos = offs[d] + atomicAdd(&cur[d], 1);
  csr_src[pos] = s;  // store source node id directly
}

// ---------------------------------------------------------------------------
// 2) SAGE softmax aggregation, one block per dst node, online softmax.
//    Channels striped over lanes -> each edge reads one contiguous row.
// ---------------------------------------------------------------------------
template <int C, int TPB>
__global__ __launch_bounds__(TPB) void k_sage_agg(
    const float* __restrict__ H, const float* __restrict__ tvec,
    const int* __restrict__ offs, const int* __restrict__ csr_src,
    float* __restrict__ agg) {
  constexpr int CPT = C / TPB;
  int node = blockIdx.x;
  int beg = offs[node], end = offs[node + 1];
  int cb = threadIdx.x;
  float m[CPT], s[CPT], num[CPT], tv[CPT];
#pragma unroll
  for (int k = 0; k < CPT; ++k) {
    m[k] = -INFINITY;
    s[k] = 0.f;
    num[k] = 0.f;
    tv[k] = tvec[cb + k * TPB];
  }
  for (int e = beg; e < end; ++e) {
    int sidx = csr_src[e];
    if (e + 1 < end) {
      // CDNA5 global_prefetch_b8 on next gathered row
      __builtin_prefetch(&H[(size_t)csr_src[e + 1] * C + cb], 0, 1);
    }
    const float* row = H + (size_t)sidx * C;
#pragma unroll
    for (int k = 0; k < CPT; ++k) {
      float hv = row[cb + k * TPB];
      float g = hv * tv[k];
      float nm = fmaxf(m[k], g);
      float scale = __expf(m[k] - nm);  // exp(-inf)=0 handles first edge
      float p = __expf(g - nm);
      s[k] = s[k] * scale + p;
      num[k] = num[k] * scale + hv * p;
      m[k] = nm;
    }
  }
  float* out = agg + (size_t)node * C;
#pragma unroll
  for (int k = 0; k < CPT; ++k)
    out[cb + k * TPB] = (s[k] > 0.f) ? (num[k] / s[k]) : 0.f;
}

// ---------------------------------------------------------------------------
// 3) Dual GEMM with WMMA f32 16x16x4:  Out = A1@B1 + A2@B2
//    A*: [M,K] row-major, B*: [K,Ncol] row-major, M%16==0, K%4==0, Ncol%32==0
//    One wave -> 16x32 output tile (two 16x16 accumulators).
// ---------------------------------------------------------------------------
__global__ __launch_bounds__(256) void k_dual_gemm_wmma(
    const float* __restrict__ A1, const float* __restrict__ B1,
    const float* __restrict__ A2, const float* __restrict__ B2,
    float* __restrict__ Out, int M, int K, int Ncol) {
  const int wavesPerBlk = 256 / 32;
  int wave = blockIdx.x * wavesPerBlk + (threadIdx.x >> 5);
  int ntiles = Ncol >> 5;              // 32-wide column tiles
  int mtiles = M >> 4;
  int mt = wave / ntiles;
  int nt = wave % ntiles;
  if (mt >= mtiles) return;

  int lane = threadIdx.x & 31;
  int half = lane >> 4;  // 0: K pair {0,1}, 1: K pair {2,3}
  int lid = lane & 15;
  int r = mt * 16 + lid;     // A row held by this lane
  int c0 = nt * 32 + lid;    // B/D column held by this lane

  v8f acc0 = {0.f, 0.f, 0.f, 0.f, 0.f, 0.f, 0.f, 0.f};
  v8f acc1 = {0.f, 0.f, 0.f, 0.f, 0.f, 0.f, 0.f, 0.f};

  const float* a1p = A1 + (size_t)r * K;
  const float* a2p = A2 + (size_t)r * K;

  for (int kk = 0; kk < K; kk += 4) {
    int ka = kk + 2 * half;
    v2f a1 = *(const v2f*)(a1p + ka);
    v2f a2 = *(const v2f*)(a2p + ka);
    const float* b1r0 = B1 + (size_t)ka * Ncol;
    const float* b2r0 = B2 + (size_t)ka * Ncol;
    v2f b1a = {b1r0[c0], b1r0[Ncol + c0]};
    v2f b1b = {b1r0[c0 + 16], b1r0[Ncol + c0 + 16]};
    v2f b2a = {b2r0[c0], b2r0[Ncol + c0]};
    v2f b2b = {b2r0[c0 + 16], b2r0[Ncol + c0 + 16]};
    acc0 = wmma_f32_4(a1, b1a, acc0);
    acc0 = wmma_f32_4(a2, b2a, acc0);
    acc1 = wmma_f32_4(a1, b1b, acc1);
    acc1 = wmma_f32_4(a2, b2b, acc1);
  }
  // C/D layout: VGPR v, lanes 0-15 -> M=v, lanes 16-31 -> M=8+v; N=lid
#pragma unroll
  for (int v = 0; v < 8; ++v) {
    int row = mt * 16 + v + 8 * half;
    Out[(size_t)row * Ncol + c0] = acc0[v];
    Out[(size_t)row * Ncol + c0 + 16] = acc1[v];
  }
}

// ---------------------------------------------------------------------------
// 4) Global (graph) LayerNorm: sum/sumsq reduce, then normalize + ReLU
// ---------------------------------------------------------------------------
__global__ __launch_bounds__(256) void k_reduce_sumsq(
    const float* __restrict__ X, size_t n, double* __restrict__ red) {
  __shared__ double ss[256], sq[256];
  double s = 0.0, q = 0.0;
  for (size_t i = (size_t)blockIdx.x * blockDim.x + threadIdx.x; i < n;
       i += (size_t)gridDim.x * blockDim.x) {
    float v = X[i];
    s += (double)v;
    q += (double)v * (double)v;
  }
  int t = threadIdx.x;
  ss[t] = s;
  sq[t] = q;
  __syncthreads();
  for (int o = 128; o > 0; o >>= 1) {
    if (t < o) {
      ss[t] += ss[t + o];
      sq[t] += sq[t + o];
    }
    __syncthreads();
  }
  if (t == 0) {
    atomicAdd(&red[0], ss[0]);
    atomicAdd(&red[1], sq[0]);
  }
}

__global__ __launch_bounds__(256) void k_graph_ln_relu(
    float* __restrict__ X, const float* __restrict__ w,
    const float* __restrict__ b, const double* __restrict__ red, size_t n,
    int cmask) {
  double mu_d = red[0] / (double)n;
  double var_d = red[1] / (double)n - mu_d * mu_d;
  float mu = (float)mu_d;
  float istd = (float)(1.0 / sqrt(var_d + 1e-5));
  for (size_t i = (size_t)blockIdx.x * blockDim.x + threadIdx.x; i < n;
       i += (size_t)gridDim.x * blockDim.x) {
    int c = (int)(i & (size_t)cmask);
    float v = (X[i] - mu) * istd * w[c] + b[c];
    X[i] = fmaxf(v, 0.f);
  }
}

// ---------------------------------------------------------------------------
// 5) MemPooling with K=1 collapses to a column sum: zp[c] = sum_n h2[n,c]
// ---------------------------------------------------------------------------
__global__ __launch_bounds__(256) void k_colsum256(const float* __restrict__ H,
                                                   float* __restrict__ zp,
                                                   int n) {
  int c = threadIdx.x;
  int rpb = (n + gridDim.x - 1) / gridDim.x;
  int r0 = blockIdx.x * rpb;
  int r1 = min(n, r0 + rpb);
  float acc = 0.f;
  for (int r = r0; r < r1; ++r) acc += H[(size_t)r * 256 + c];
  atomicAdd(&zp[c], acc);
}

// ---------------------------------------------------------------------------
// 6) z = (zp @ mp_lin) @ fxg_w + fxg_b ;  q = softmax(z@fk+bk) ; v = z@fv+bv
//    qv layout: qv[0..63] = q, qv[64..127] = v
// ---------------------------------------------------------------------------
__global__ __launch_bounds__(256) void k_head_zqv(
    const float* __restrict__ zp, const float* __restrict__ mp_lin,
    const float* __restrict__ fxg_w, const float* __restrict__ fxg_b,
    const float* __restrict__ fk_w, const float* __restrict__ fk_b,
    const float* __restrict__ fv_w, const float* __restrict__ fv_b,
    float* __restrict__ qv) {
  __shared__ float zs[256], tm[128], z[256], ql[64];
  int t = threadIdx.x;
  zs[t] = zp[t];
  __syncthreads();
  if (t < 128) {
    float a = 0.f;
    for (int i = 0; i < 256; ++i) a += zs[i] * mp_lin[i * 128 + t];
    tm[t] = a;
  }
  __syncthreads();
  {
    float a = fxg_b[t];
    for (int i = 0; i < 128; ++i) a += tm[i] * fxg_w[i * 256 + t];
    z[t] = a;
  }
  __syncthreads();
  if (t < 64) {
    float a = fk_b[t];
    for (int i = 0; i < 256; ++i) a += z[i] * fk_w[i * 64 + t];
    ql[t] = a;
  } else if (t < 128) {
    int j = t - 64;
    float a = fv_b[j];
    for (int i = 0; i < 256; ++i) a += z[i] * fv_w[i * 64 + j];
    qv[64 + j] = a;
  }
  __syncthreads();
  if (t == 0) {
    float mx = -INFINITY;
    for (int i = 0; i < 64; ++i) mx = fmaxf(mx, ql[i]);
    float se = 0.f;
    for (int i = 0; i < 64; ++i) {
      float e = __expf(ql[i] - mx);
      qv[i] = e;
      se += e;
    }
    float inv = 1.f / se;
    for (int i = 0; i < 64; ++i) qv[i] *= inv;
  }
}

// ---------------------------------------------------------------------------
// 7) Per-memory-row: logit = q . softmax(row@mfk_w+b), vm row = row@mfv_w+b
// ---------------------------------------------------------------------------
__global__ __launch_bounds__(64) void k_mem_kv(
    const float* __restrict__ mem, const float* __restrict__ mfk_w,
    const float* __restrict__ mfk_b, const float* __restrict__ mfv_w,
    const float* __restrict__ mfv_b, const float* __restrict__ qv,
    float* __restrict__ logits, float* __restrict__ vm) {
  int r = blockIdx.x * blockDim.x + threadIdx.x;  // 0..4095 = h*1024+n
  if (r >= 4096) return;
  const float* row = mem + (size_t)r * 128;
  float acc[64];
#pragma unroll
  for (int j = 0; j < 64; ++j) acc[j] = mfk_b[j];
  for (int d = 0; d < 128; ++d) {
    float rv = row[d];
    const float* wr = mfk_w + d * 64;
#pragma unroll
    for (int j = 0; j < 64; ++j) acc[j] += rv * wr[j];
  }
  float mx = acc[0];
#pragma unroll
  for (int j = 1; j < 64; ++j) mx = fmaxf(mx, acc[j]);
  float se = 0.f, dot = 0.f;
#pragma unroll
  for (int j = 0; j < 64; ++j) {
    float e = __expf(acc[j] - mx);
    se += e;
    dot += qv[j] * e;
  }
  logits[r] = dot / se;
#pragma unroll
  for (int j = 0; j < 64; ++j) acc[j] = mfv_b[j];
  for (int d = 0; d < 128; ++d) {
    float rv = row[d];
    const float* wr = mfv_w + d * 64;
#pragma unroll
    for (int j = 0; j < 64; ++j) acc[j] += rv * wr[j];
  }
  float* vout = vm + (size_t)r * 64;
#pragma unroll
  for (int j = 0; j < 64; ++j) vout[j] = acc[j];
}

// ---------------------------------------------------------------------------
// 8) Per-head softmax over 1024 memory slots + weighted sum of vm
// ---------------------------------------------------------------------------
__global__ __launch_bounds__(256) void k_mem_attend(
    const float* __restrict__ logits, const float* __restrict__ vm,
    float* __restrict__ mv) {
  int h = blockIdx.x, t = threadIdx.x;
  __shared__ float red[256];
  __shared__ float wl[1024];
  const float* lg = logits + h * 1024;
  float mx = -INFINITY;
  for (int i = t; i < 1024; i += 256) mx = fmaxf(mx, lg[i]);
  red[t] = mx;
  __syncthreads();
  for (int o = 128; o > 0; o >>= 1) {
    if (t < o) red[t] = fmaxf(red[t], red[t + o]);
    __syncthreads();
  }
  mx = red[0];
  __syncthreads();
  float s = 0.f;
  for (int i = t; i < 1024; i += 256) {
    float e = __expf(lg[i] - mx);
    wl[i] = e;
    s += e;
  }
  red[t] = s;
  __syncthreads();
  for (int o = 128; o > 0; o >>= 1) {
    if (t < o) red[t] += red[t + o];
    __syncthreads();
  }
  float inv = 1.f / red[0];
  __syncthreads();
  if (t < 64) {
    float acc = 0.f;
    const float* vb = vm + (size_t)h * 1024 * 64 + t;
    for (int n2 = 0; n2 < 1024; ++n2) acc += wl[n2] * vb[(size_t)n2 * 64];
    mv[h * 64 + t] = acc * inv;
  }
}

__global__ __launch_bounds__(64) void k_mem_out(const float* __restrict__ mv,
                                                const float* __restrict__ mfx_w,
                                                const float* __restrict__ mfx_b,
                                                float* __restrict__ mout) {
  int j = threadIdx.x;
  float a = mfx_b[j];
  for (int i = 0; i < 256; ++i) a += mv[i] * mfx_w[i * 64 + j];
  mout[j] = a;
}

// ---------------------------------------------------------------------------
// 9) Final per-type MLP: feat[273] -> 512 -> 256 -> 3, LN + ReLU, softmax
// ---------------------------------------------------------------------------
__global__ __launch_bounds__(256) void k_policy_head(
    const float* __restrict__ x, const int* __restrict__ index_p,
    const int* __restrict__ inp_types, const float* __restrict__ log_return,
    const float* __restrict__ emb, const float* __restrict__ qv,
    const float* __restrict__ mout, const float* __restrict__ f1_w,
    const float* __restrict__ n1_w, const float* __restrict__ n1_b,
    const float* __restrict__ f2_w, const float* __restrict__ n2_w,
    const float* __restrict__ n2_b, const float* __restrict__ f3_w,
    const float* __restrict__ f3_b, float* __restrict__ out) {
  int trow = blockIdx.x, t = threadIdx.x;
  __shared__ float feat[273];
  __shared__ float h1[512];
  __shared__ float h2[256];
  __shared__ float red[256];
  int idx = index_p[0];
  int ity = inp_types[trow];
  for (int i = t; i < 273; i += 256) {
    float fv;
    if (i < 128) fv = x[(size_t)idx * 128 + i];
    else if (i < 144) fv = emb[ity * 16 + (i - 128)];
    else if (i == 144) fv = log_return[trow];
    else if (i < 209) fv = qv[64 + (i - 145)];  // v
    else fv = mout[i - 209];                    // m_out
    feat[i] = fv;
  }
  __syncthreads();
  for (int j = t; j < 512; j += 256) {
    float a = 0.f;
    for (int i = 0; i < 273; ++i) a += feat[i] * f1_w[i * 512 + j];
    h1[j] = a;
  }
  __syncthreads();
  // LN(512) + ReLU
  red[t] = h1[t] + h1[t + 256];
  __syncthreads();
  for (int o = 128; o > 0; o >>= 1) {
    if (t < o) red[t] += red[t + o];
    __syncthreads();
  }
  float mu = red[0] / 512.f;
  __syncthreads();
  {
    float d0 = h1[t] - mu, d1 = h1[t + 256] - mu;
    red[t] = d0 * d0 + d1 * d1;
  }
  __syncthreads();
  for (int o = 128; o > 0; o >>= 1) {
    if (t < o) red[t] += red[t + o];
    __syncthreads();
  }
  float istd = rsqrtf(red[0] / 512.f + 1e-5f);
  __syncthreads();
  h1[t] = fmaxf(0.f, (h1[t] - mu) * istd * n1_w[t] + n1_b[t]);
  h1[t + 256] =
      fmaxf(0.f, (h1[t + 256] - mu) * istd * n1_w[t + 256] + n1_b[t + 256]);
  __syncthreads();
  {
    float a = 0.f;
    for (int i = 0; i < 512; ++i) a += h1[i] * f2_w[i * 256 + t];
    h2[t] = a;
  }
  __syncthreads();
  // LN(256) + ReLU
  red[t] = h2[t];
  __syncthreads();
  for (int o = 128; o > 0; o >>= 1) {
    if (t < o) red[t] += red[t + o];
    __syncthreads();
  }
  mu = red[0] / 256.f;
  __syncthreads();
  {
    float d = h2[t] - mu;
    red[t] = d * d;
  }
  __syncthreads();
  for (int o = 128; o > 0; o >>= 1) {
    if (t < o) red[t] += red[t + o];
    __syncthreads();
  }
  istd = rsqrtf(red[0] / 256.f + 1e-5f);
  __syncthreads();
  h2[t] = fmaxf(0.f, (h2[t] - mu) * istd * n2_w[t] + n2_b[t]);
  __syncthreads();
  if (t < 3) {
    float a = f3_b[t];
    for (int i = 0; i < 256; ++i) a += h2[i] * f3_w[i * 3 + t];
    red[t] = a;
  }
  __syncthreads();
  if (t == 0) {
    float mx = fmaxf(red[0], fmaxf(red[1], red[2]));
    float e0 = __expf(red[0] - mx), e1 = __expf(red[1] - mx),
          e2 = __expf(red[2] - mx);
    float inv = 1.f / (e0 + e1 + e2);
    out[trow * 3 + 0] = e0 * inv;
    out[trow * 3 + 1] = e1 * inv;
    out[trow * 3 + 2] = e2 * inv;
  }
}

// ---------------------------------------------------------------------------
// Launch
// ---------------------------------------------------------------------------
static inline size_t al256(size_t x) { return (x + 255) & ~(size_t)255; }

extern "C" void kernel_launch(void* const* d_in, const int* in_sizes, int n_in,
                              void* d_out, int out_size, void* d_ws,
                              size_t ws_size, hipStream_t stream) {
  (void)n_in;
  (void)out_size;
  (void)ws_size;
  const float* x = (const float*)d_in[0];
  const int* ei = (const int*)d_in[1];
  const int* index_p = (const int*)d_in[2];
  const int* inp_types = (const int*)d_in[3];
  const float* log_return = (const float*)d_in[4];
  const float* t1 = (const float*)d_in[5];
  const float* W1l = (const float*)d_in[6];
  const float* W1r = (const float*)d_in[7];
  const float* ln1g_w = (const float*)d_in[8];
  const float* ln1g_b = (const float*)d_in[9];
  const float* t2 = (const float*)d_in[10];
  const float* W2l = (const float*)d_in[11];
  const float* W2r = (const float*)d_in[12];
  const float* ln2g_w = (const float*)d_in[13];
  const float* ln2g_b = (const float*)d_in[14];
  // d_in[15] mp_k, d_in[16] mp_conv: drop out algebraically (K==1 clusters)
  const float* mp_lin = (const float*)d_in[17];
  const float* fxg_w = (const float*)d_in[18];
  const float* fxg_b = (const float*)d_in[19];
  const float* emb = (const float*)d_in[20];
  const float* fk_w = (const float*)d_in[21];
  const float* fk_b = (const float*)d_in[22];
  const float* fv_w = (const float*)d_in[23];
  const float* fv_b = (const float*)d_in[24];
  const float* mem = (const float*)d_in[25];
  const float* mfk_w = (const float*)d_in[26];
  const float* mfk_b = (const float*)d_in[27];
  const float* mfv_w = (const float*)d_in[28];
  const float* mfv_b = (const float*)d_in[29];
  const float* mfx_w = (const float*)d_in[30];
  const float* mfx_b = (const float*)d_in[31];
  const float* f1_w = (const float*)d_in[32];
  const float* n1_w = (const float*)d_in[33];
  const float* n1_b = (const float*)d_in[34];
  const float* f2_w = (const float*)d_in[35];
  const float* n2_w = (const float*)d_in[36];
  const float* n2_b = (const float*)d_in[37];
  const float* f3_w = (const float*)d_in[38];
  const float* f3_b = (const float*)d_in[39];

  const int N = in_sizes[0] / 128;  // 20000
  const int E = in_sizes[1] / 2;    // 320000
  const int T = in_sizes[3];        // 64

  // ---- workspace layout ----
  char* w = (char*)d_ws;
  size_t off = 0;
  auto wsalloc = [&](size_t bytes) -> void* {
    void* p = (void*)(w + off);
    off = al256(off + bytes);
    return p;
  };
  int* deg = (int*)wsalloc((size_t)N * 4);
  int* offs = (int*)wsalloc((size_t)(N + 1) * 4);
  int* cur = (int*)wsalloc((size_t)N * 4);
  int* csr = (int*)wsalloc((size_t)E * 4);
  // agg1 [N,128] and h2 [N,256] share one slot (agg1 dead before h2 written)
  float* agg1 = (float*)wsalloc((size_t)N * 256 * 4);
  float* h2 = agg1;
  float* h1 = (float*)wsalloc((size_t)N * 512 * 4);
  float* agg2 = (float*)wsalloc((size_t)N * 512 * 4);
  double* red = (double*)wsalloc(4 * sizeof(double));
  float* zp = (float*)wsalloc(256 * 4);
  float* qv = (float*)wsalloc(128 * 4);
  float* logits = (float*)wsalloc(4096 * 4);
  float* vm = (float*)wsalloc((size_t)4096 * 64 * 4);
  float* mv = (float*)wsalloc(256 * 4);
  float* mout = (float*)wsalloc(64 * 4);

  // zero accumulators (capturable async memsets)
  hipMemsetAsync(deg, 0, (size_t)N * 4, stream);
  hipMemsetAsync(cur, 0, (size_t)N * 4, stream);
  hipMemsetAsync(red, 0, 4 * sizeof(double), stream);
  hipMemsetAsync(zp, 0, 256 * 4, stream);

  // CSR build
  k_count_deg<<<(E + 255) / 256, 256, 0, stream>>>(ei, deg, E);
  k_scan<<<1, 1024, 0, stream>>>(deg, offs, N);
  k_scatter<<<(E + 255) / 256, 256, 0, stream>>>(ei, offs, cur, csr, E);

  // Layer 1
  k_sage_agg<128, 128><<<N, 128, 0, stream>>>(x, t1, offs, csr, agg1);
  {
    int mtiles = N / 16, ntiles = 512 / 32;
    int waves = mtiles * ntiles;
    k_dual_gemm_wmma<<<(waves + 7) / 8, 256, 0, stream>>>(agg1, W1l, x, W1r, h1,
                                                          N, 128, 512);
  }
  k_reduce_sumsq<<<1024, 256, 0, stream>>>(h1, (size_t)N * 512, red);
  k_graph_ln_relu<<<2048, 256, 0, stream>>>(h1, ln1g_w, ln1g_b, red,
                                            (size_t)N * 512, 511);

  // Layer 2
  k_sage_agg<512, 256><<<N, 256, 0, stream>>>(h1, t2, offs, csr, agg2);
  {
    int mtiles = N / 16, ntiles = 256 / 32;
    int waves = mtiles * ntiles;
    k_dual_gemm_wmma<<<(waves + 7) / 8, 256, 0, stream>>>(agg2, W2l, h1, W2r,
                                                          h2, N, 512, 256);
  }
  k_reduce_sumsq<<<1024, 256, 0, stream>>>(h2, (size_t)N * 256, red + 2);
  k_graph_ln_relu<<<2048, 256, 0, stream>>>(h2, ln2g_w, ln2g_b, red + 2,
                                            (size_t)N * 256, 255);

  // MemPooling (K=1) -> column sum, then head
  k_colsum256<<<200, 256, 0, stream>>>(h2, zp, N);
  k_head_zqv<<<1, 256, 0, stream>>>(zp, mp_lin, fxg_w, fxg_b, fk_w, fk_b, fv_w,
                                    fv_b, qv);
  k_mem_kv<<<64, 64, 0, stream>>>(mem, mfk_w, mfk_b, mfv_w, mfv_b, qv, logits,
                                  vm);
  k_mem_attend<<<4, 256, 0, stream>>>(logits, vm, mv);
  k_mem_out<<<1, 64, 0, stream>>>(mv, mfx_w, mfx_b, mout);

  // Final MLP over T rows
  k_policy_head<<<T, 256, 0, stream>>>(x, index_p, inp_types, log_return, emb,
                                       qv, mout, f1_w, n1_w, n1_b, f2_w, n2_w,
                                       n2_b, f3_w, f3_b, (float*)d_out);
}